// SimpleGNN_43654047596746
// MI455X (gfx1250) — compile-verified
//
#include <hip/hip_runtime.h>
#include <hip/hip_bf16.h>

#define HID 64
#define BN_EPS 1e-5f

typedef __attribute__((ext_vector_type(2))) float v2f;
typedef __attribute__((ext_vector_type(8))) float v8f;
typedef __attribute__((ext_vector_type(4))) unsigned int u32x4;
typedef __attribute__((ext_vector_type(8))) int i32x8;
typedef __attribute__((ext_vector_type(4))) int i32x4;

// ---------------------------------------------------------------------------
// Zero fill
// ---------------------------------------------------------------------------
__global__ void zero_f32(float* __restrict__ p, int n) {
  int i = blockIdx.x * blockDim.x + threadIdx.x;
  if (i < n) p[i] = 0.0f;
}

// ---------------------------------------------------------------------------
// GIN neighbor aggregation: agg[dst] += h[src] over all edges.
// One thread handles 4 consecutive features of one edge (float4 gather,
// 4x global_atomic_add_f32 scatter).
// ---------------------------------------------------------------------------
__global__ void gin_aggregate(const float* __restrict__ h, const int* __restrict__ src,
                              const int* __restrict__ dst, float* __restrict__ agg, int e) {
  int tid = blockIdx.x * blockDim.x + threadIdx.x;
  int total = e * 16;
  if (tid >= total) return;
  int edge = tid >> 4;
  int f = (tid & 15) << 2;
  int s = src[edge];
  int d = dst[edge];
  const float4 hv = *(const float4*)(h + (size_t)s * HID + f);
  float* o = agg + (size_t)d * HID + f;
  atomicAdd(o + 0, hv.x);
  atomicAdd(o + 1, hv.y);
  atomicAdd(o + 2, hv.z);
  atomicAdd(o + 3, hv.w);
}

// ---------------------------------------------------------------------------
// Issue a TDM DMA: copy `elems` fp32 values from global `W` into LDS at byte
// offset `lds_off` as one 1-D tile (elems <= 65535 fits 16-bit tile_dim0).
// D# layout per CDNA5 ISA ch. 8 (groups 0/1; groups 2/3 unused -> zero).
// Must be executed by a single wave; caller waits TENSORcnt + barrier.
// ---------------------------------------------------------------------------
__device__ __forceinline__ void tdm_load_w_to_lds(const float* W, unsigned lds_off,
                                                  unsigned elems) {
  unsigned long long ga = (unsigned long long)W;
  u32x4 g0;
  g0.x = 1u;                                              // count=1, user mode
  g0.y = lds_off;                                         // lds_addr (bytes)
  g0.z = (unsigned)(ga & 0xFFFFFFFFu);                    // global_addr[31:0]
  g0.w = (unsigned)((ga >> 32) & 0x01FFFFFFu) | (2u << 30);  // addr[56:32] | type=2

  i32x8 g1;
  g1[0] = 0x00020000;                                     // data_size=2 (4 bytes)
  g1[1] = (int)((elems & 0xFFFFu) << 16);                 // tensor_dim0[15:0]
  g1[2] = (int)(((elems >> 16) & 0xFFFFu) | (1u << 16));  // dim0[31:16] | tensor_dim1=1
  g1[3] = (int)((elems & 0xFFFFu) << 16);                 // tile_dim0 = elems
  g1[4] = 1;                                              // tile_dim1=1, tile_dim2=0
  g1[5] = (int)elems;                                     // tensor_dim0_stride[31:0]
  g1[6] = 0;                                              // stride hi / dim1_stride lo
  g1[7] = 0;

  i32x4 gz = {0, 0, 0, 0};
#if defined(__clang_major__) && (__clang_major__ >= 23)
  i32x8 gz8 = {0, 0, 0, 0, 0, 0, 0, 0};
  __builtin_amdgcn_tensor_load_to_lds(g0, g1, gz, gz, gz8, 0);
#else
  __builtin_amdgcn_tensor_load_to_lds(g0, g1, gz, gz, 0);
#endif
}

// ---------------------------------------------------------------------------
// Fused GEMM (+ optional elementwise add of a second input) + bias + eval-BN
// + ReLU, using V_WMMA_F32_16X16X4_F32. Weights streamed into LDS by the
// Tensor Data Mover (tensor_load_to_lds + s_wait_tensorcnt).
//
//   out[n x 64] = relu( BN( (A (+A2)) @ W + bias ) )
//
// fp32 WMMA fragment layouts (CDNA5 ISA 7.12.2):
//   A 16x4 : lane = khalf*16 + m (m=lane&15), vgpr v holds K = 2*khalf + v
//   B 4x16 : lane = khalf*16 + n, vgpr v holds K = 2*khalf + v
//   C 16x16: lane&15 = N, vgpr v holds M = v + 8*(lane>>4)
// ---------------------------------------------------------------------------
template <int K, bool HASADD>
__global__ void gemm_bn_relu(const float* __restrict__ A, const float* __restrict__ A2,
                             const float* __restrict__ W, const float* __restrict__ bias,
                             const float* __restrict__ bng, const float* __restrict__ bnb,
                             const float* __restrict__ bnm, const float* __restrict__ bnv,
                             float* __restrict__ out, int nrows) {
  __shared__ float sW[K * HID];

  // Wave 0 issues one TDM descriptor covering the whole K x 64 weight block,
  // waits on TENSORcnt; the workgroup barrier releases the other waves.
  if (threadIdx.x < 32) {
    unsigned lds_off = (unsigned)(unsigned long long)(void*)&sW[0];
    tdm_load_w_to_lds(W, lds_off, (unsigned)(K * HID));
    __builtin_amdgcn_s_wait_tensorcnt(0);
  }
  __syncthreads();

  const int lane  = threadIdx.x & 31;
  const int wave  = threadIdx.x >> 5;
  const int lrow  = lane & 15;
  const int lhalf = lane >> 4;

  const int m0 = blockIdx.x * 32 + (wave & 1) * 16;
  const int n0 = (wave >> 1) * 16;

  const int row   = m0 + lrow;
  const int rsafe = row < nrows ? row : nrows - 1;
  const float* arow  = A + (size_t)rsafe * K;
  const float* a2row = HASADD ? (A2 + (size_t)rsafe * K) : nullptr;

  v8f acc = {};
#pragma unroll
  for (int k = 0; k < K; k += 4) {
    const int ka = k + lhalf * 2;
    v2f a, b;
    if (HASADD) {
      a.x = arow[ka + 0] + a2row[ka + 0];
      a.y = arow[ka + 1] + a2row[ka + 1];
    } else {
      a.x = arow[ka + 0];
      a.y = arow[ka + 1];
    }
    b.x = sW[(ka + 0) * HID + n0 + lrow];
    b.y = sW[(ka + 1) * HID + n0 + lrow];
    acc = __builtin_amdgcn_wmma_f32_16x16x4_f32(
        /*neg_a=*/false, a, /*neg_b=*/false, b,
        /*c_mod=*/(short)0, acc, /*reuse_a=*/false, /*reuse_b=*/false);
  }

  // Epilogue: relu(((acc + bias) - m) * g*rsqrt(v+eps) + b) per output column.
  const int ncol = n0 + lrow;
  const float scale = bng[ncol] * rsqrtf(bnv[ncol] + BN_EPS);
  const float shift = bnb[ncol] + (bias[ncol] - bnm[ncol]) * scale;
#pragma unroll
  for (int v = 0; v < 8; ++v) {
    const int m = m0 + lhalf * 8 + v;
    if (m < nrows) {
      float val = acc[v] * scale + shift;
      out[(size_t)m * HID + ncol] = val > 0.0f ? val : 0.0f;
    }
  }
}

// ---------------------------------------------------------------------------
// Global mean-pool phase 1: per-graph feature sums + node counts.
// ---------------------------------------------------------------------------
__global__ void pool_sum(const float* __restrict__ h, const int* __restrict__ batch,
                         float* __restrict__ sums, float* __restrict__ cnts, int n) {
  int tid = blockIdx.x * blockDim.x + threadIdx.x;
  int total = n * 16;
  if (tid >= total) return;
  int node = tid >> 4;
  int f = (tid & 15) << 2;
  int g = batch[node];
  const float4 hv = *(const float4*)(h + (size_t)node * HID + f);
  float* o = sums + (size_t)g * HID + f;
  atomicAdd(o + 0, hv.x);
  atomicAdd(o + 1, hv.y);
  atomicAdd(o + 2, hv.z);
  atomicAdd(o + 3, hv.w);
  if ((tid & 15) == 0) atomicAdd(cnts + g, 1.0f);
}

// ---------------------------------------------------------------------------
// Classifier head: pooled = sums/max(cnt,1); relu(pooled@W1+b1) @ W2 + b2.
// One 64-thread block per graph (G=1024, negligible work).
// ---------------------------------------------------------------------------
__global__ void classifier_head(const float* __restrict__ sums, const float* __restrict__ cnts,
                                const float* __restrict__ w1, const float* __restrict__ b1,
                                const float* __restrict__ w2, const float* __restrict__ b2,
                                float* __restrict__ out) {
  __shared__ float pooled[HID];
  __shared__ float hid[HID / 2];
  int g = blockIdx.x;
  int t = threadIdx.x;
  float c = cnts[g];
  c = c > 1.0f ? c : 1.0f;
  pooled[t] = sums[(size_t)g * HID + t] / c;
  __syncthreads();
  if (t < HID / 2) {
    float acc = b1[t];
#pragma unroll 8
    for (int k = 0; k < HID; ++k) acc += pooled[k] * w1[k * (HID / 2) + t];
    hid[t] = acc > 0.0f ? acc : 0.0f;
  }
  __syncthreads();
  if (t < 6) {
    float acc = b2[t];
#pragma unroll 8
    for (int k = 0; k < HID / 2; ++k) acc += hid[k] * w2[k * 6 + t];
    out[g * 6 + t] = acc;
  }
}

// ---------------------------------------------------------------------------
// Launcher
// ---------------------------------------------------------------------------
extern "C" void kernel_launch(void* const* d_in, const int* in_sizes, int n_in,
                              void* d_out, int out_size, void* d_ws, size_t ws_size,
                              hipStream_t stream) {
  (void)n_in; (void)ws_size;
  const float* x      = (const float*)d_in[0];
  const int*   edge   = (const int*)d_in[1];
  const int*   batch  = (const int*)d_in[2];
  const float* emb_w  = (const float*)d_in[3];
  const float* emb_b  = (const float*)d_in[4];
  const float* ibn_g  = (const float*)d_in[5];
  const float* ibn_b  = (const float*)d_in[6];
  const float* ibn_m  = (const float*)d_in[7];
  const float* ibn_v  = (const float*)d_in[8];
  const float* fc1_w  = (const float*)d_in[9];
  const float* fc1_b  = (const float*)d_in[10];
  const float* mbn_g  = (const float*)d_in[11];
  const float* mbn_b  = (const float*)d_in[12];
  const float* mbn_m  = (const float*)d_in[13];
  const float* mbn_v  = (const float*)d_in[14];
  const float* fc2_w  = (const float*)d_in[15];
  const float* fc2_b  = (const float*)d_in[16];
  const float* obn_g  = (const float*)d_in[17];
  const float* obn_b  = (const float*)d_in[18];
  const float* obn_m  = (const float*)d_in[19];
  const float* obn_v  = (const float*)d_in[20];
  const float* cls1_w = (const float*)d_in[21];
  const float* cls1_b = (const float*)d_in[22];
  const float* cls2_w = (const float*)d_in[23];
  const float* cls2_b = (const float*)d_in[24];

  const int n      = in_sizes[0] / 128;  // N nodes
  const int e      = in_sizes[1] / 2;    // E edges
  const int ngraph = out_size / 6;       // G graphs

  // Workspace layout (floats): h | agg | z | sums | cnts
  float* h    = (float*)d_ws;
  float* agg  = h + (size_t)n * HID;
  float* z    = agg + (size_t)n * HID;
  float* sums = z + (size_t)n * HID;
  float* cnts = sums + (size_t)ngraph * HID;

  const int gemmGrid = (n + 31) / 32;
  const int feats    = n * HID;

  // Input embedding + input BN + ReLU
  gemm_bn_relu<128, false><<<gemmGrid, 256, 0, stream>>>(
      x, nullptr, emb_w, emb_b, ibn_g, ibn_b, ibn_m, ibn_v, h, n);

  // 3 GIN layers
  for (int i = 0; i < 3; ++i) {
    zero_f32<<<(feats + 255) / 256, 256, 0, stream>>>(agg, feats);
    gin_aggregate<<<(e * 16 + 255) / 256, 256, 0, stream>>>(h, edge, edge + e, agg, e);
    gemm_bn_relu<64, true><<<gemmGrid, 256, 0, stream>>>(
        h, agg, fc1_w + (size_t)i * HID * HID, fc1_b + i * HID,
        mbn_g + i * HID, mbn_b + i * HID, mbn_m + i * HID, mbn_v + i * HID, z, n);
    gemm_bn_relu<64, false><<<gemmGrid, 256, 0, stream>>>(
        z, nullptr, fc2_w + (size_t)i * HID * HID, fc2_b + i * HID,
        obn_g + i * HID, obn_b + i * HID, obn_m + i * HID, obn_v + i * HID, h, n);
  }

  // Mean pool + classifier
  const int pn = ngraph * HID + ngraph;  // sums and cnts are contiguous
  zero_f32<<<(pn + 255) / 256, 256, 0, stream>>>(sums, pn);
  pool_sum<<<(n * 16 + 255) / 256, 256, 0, stream>>>(h, batch, sums, cnts, n);
  classifier_head<<<ngraph, 64, 0, stream>>>(sums, cnts, cls1_w, cls1_b, cls2_w, cls2_b,
                                             (float*)d_out);
}